// MedianPool_73426760893099
// MI455X (gfx1250) — compile-verified
//
#include <hip/hip_runtime.h>
#include <hip/hip_bf16.h>

// MedianPool: x [B=32, L=4096, C=128] fp32, circular sliding median along L,
// window w (=5 in the reference setup). out[b,l,c] = lower-median of
// { x[b,(l+i) mod L, c] : i in [0,w) }.
//
// Memory-bound: 64 MiB in + 64 MiB out -> ~5.5us floor at 23.3 TB/s.
// Strategy: thread = (b, strip-of-16 l values, group of 4 channels).
// Rolling 5-deep register window => ~1 float4 load + 1 float4 store per
// output row step, fully coalesced B128 accesses, global_prefetch ahead.
// Median-of-5 via v_med3_f32: med5 = med3(e, max(min(a,b),min(c,d)),
//                                            min(max(a,b),max(c,d)))  (5 ops).

#define B_DIM   32
#define L_DIM   4096
#define C_DIM   128
#define C4_DIM  (C_DIM / 4)     // 32 float4 per row
#define LMASK   (L_DIM - 1)
#define L_PER_THREAD 16
#define SUBS_PER_BLOCK 8        // 256 threads = 32 c4-lanes x 8 l-substrips
#define L_PER_BLOCK (L_PER_THREAD * SUBS_PER_BLOCK)   // 128
#define STRIPS  (L_DIM / L_PER_BLOCK)                 // 32

// med5 = med3(e, 2nd-smallest{a,b,c,d}, 2nd-largest{a,b,c,d})
// For any pairing of 4 values, {max(pair minima), min(pair maxima)} = {x2, x3}.
__device__ __forceinline__ float med5f(float a, float b, float c, float d, float e) {
    float lo01 = fminf(a, b), hi01 = fmaxf(a, b);
    float lo23 = fminf(c, d), hi23 = fmaxf(c, d);
    float m1 = fmaxf(lo01, lo23);     // x2 or x3 of the four
    float m2 = fminf(hi01, hi23);     // the other of x2, x3
    return __builtin_amdgcn_fmed3f(e, m1, m2);   // v_med3_f32
}

__global__ __launch_bounds__(256) void median_pool_kernel(
    const float* __restrict__ xf,
    const int*   __restrict__ wptr,
    float*       __restrict__ outf)
{
    const int w = *wptr;   // uniform scalar load

    const float4* __restrict__ x   = (const float4*)xf;
    float4*       __restrict__ out = (float4*)outf;

    const int b     = (int)(blockIdx.x >> 5);       // blockIdx.x / STRIPS
    const int strip = (int)(blockIdx.x & (STRIPS - 1));
    const int sub   = (int)(threadIdx.x >> 5);      // 0..7
    const int c4    = (int)(threadIdx.x & (C4_DIM - 1));
    const int l0    = strip * L_PER_BLOCK + sub * L_PER_THREAD;

    const size_t base = (size_t)b * (size_t)(L_DIM * C4_DIM) + (size_t)c4;

    if (w == 5) {
        // Rolling register window of 5 float4 rows.
        float4 v0 = x[base + (size_t)(( l0    ) & LMASK) * C4_DIM];
        float4 v1 = x[base + (size_t)((l0 + 1) & LMASK) * C4_DIM];
        float4 v2 = x[base + (size_t)((l0 + 2) & LMASK) * C4_DIM];
        float4 v3 = x[base + (size_t)((l0 + 3) & LMASK) * C4_DIM];

#pragma unroll
        for (int i = 0; i < L_PER_THREAD; ++i) {
            const int l = l0 + i;
            // CDNA5 global_prefetch_b8: pull the row ~12 steps ahead toward L2/L0.
            __builtin_prefetch((const void*)&x[base + (size_t)((l + 12) & LMASK) * C4_DIM], 0, 3);

            float4 v4 = x[base + (size_t)((l + 4) & LMASK) * C4_DIM];

            float4 m;
            m.x = med5f(v0.x, v1.x, v2.x, v3.x, v4.x);
            m.y = med5f(v0.y, v1.y, v2.y, v3.y, v4.y);
            m.z = med5f(v0.z, v1.z, v2.z, v3.z, v4.z);
            m.w = med5f(v0.w, v1.w, v2.w, v3.w, v4.w);

            out[base + (size_t)l * C4_DIM] = m;

            v0 = v1; v1 = v2; v2 = v3; v3 = v4;
        }
    } else {
        // Generic fallback (any w >= 1): scratch-free O(w^2) rank counting.
        // Lower median = element with stable rank (w-1)/2.
        const int m_idx = (w - 1) >> 1;
        for (int i = 0; i < L_PER_THREAD; ++i) {
            const int l = l0 + i;
            float4 med = make_float4(0.f, 0.f, 0.f, 0.f);
            for (int j = 0; j < w; ++j) {
                float4 vj = x[base + (size_t)((l + j) & LMASK) * C4_DIM];
                int cx = 0, cy = 0, cz = 0, cw = 0;
                for (int k = 0; k < w; ++k) {
                    float4 vk = x[base + (size_t)((l + k) & LMASK) * C4_DIM];
                    cx += (vk.x < vj.x) || (vk.x == vj.x && k < j);
                    cy += (vk.y < vj.y) || (vk.y == vj.y && k < j);
                    cz += (vk.z < vj.z) || (vk.z == vj.z && k < j);
                    cw += (vk.w < vj.w) || (vk.w == vj.w && k < j);
                }
                if (cx == m_idx) med.x = vj.x;
                if (cy == m_idx) med.y = vj.y;
                if (cz == m_idx) med.z = vj.z;
                if (cw == m_idx) med.w = vj.w;
            }
            out[base + (size_t)l * C4_DIM] = med;
        }
    }
}

extern "C" void kernel_launch(void* const* d_in, const int* in_sizes, int n_in,
                              void* d_out, int out_size, void* d_ws, size_t ws_size,
                              hipStream_t stream) {
    const float* x    = (const float*)d_in[0];
    const int*   wptr = (const int*)d_in[1];
    float*       out  = (float*)d_out;

    dim3 grid(B_DIM * STRIPS);   // 1024 blocks
    dim3 block(256);             // 8 wave32 per block
    median_pool_kernel<<<grid, block, 0, stream>>>(x, wptr, out);
}